// Coupling_28913719837465
// MI455X (gfx1250) — compile-verified
//
#include <hip/hip_runtime.h>

// Problem constants (match reference)
#define N_ALPHA 36
#define N_J     10
#define SOAP    256
#define D_J     4
#define N_PAD   16   // pad N_J -> 16 for WMMA
#define K_STEPS (N_ALPHA / 4)  // 9 K-steps of 4 for V_WMMA_F32_16X16X4_F32

typedef float v2f __attribute__((ext_vector_type(2)));
typedef float v8f __attribute__((ext_vector_type(8)));

// Kernel 1: build padded transition matrix P_pad[36][16] in workspace.
// P[a, j] = U[alpha1[a], j1[j]] * U[alpha2[a], j2[j]],  zero for j >= 10.
__global__ void build_P_kernel(const float* __restrict__ U,
                               const int* __restrict__ alpha1,
                               const int* __restrict__ alpha2,
                               const int* __restrict__ j1,
                               const int* __restrict__ j2,
                               float* __restrict__ Ppad) {
    int idx = threadIdx.x;             // 0 .. 36*16-1
    if (idx >= N_ALPHA * N_PAD) return;
    int a = idx / N_PAD;
    int n = idx % N_PAD;
    float v = 0.0f;
    if (n < N_J) {
        v = U[alpha1[a] * D_J + j1[n]] * U[alpha2[a] * D_J + j2[n]];
    }
    Ppad[idx] = v;
}

// Kernel 2: per environment GEMM [256 x 36] @ [36 x 10] via V_WMMA_F32_16X16X4_F32.
// One block (256 threads = 8 waves) per env; each wave owns 2 M-tiles of 16 soap rows.
__global__ __launch_bounds__(256)
void coupling_wmma_kernel(const float* __restrict__ F,     // [env, 36*256] stored [a][s]
                          const float* __restrict__ Ppad,  // [36][16]
                          float* __restrict__ out)         // [env, 256*10]
{
    const int e    = blockIdx.x;
    const int lane = threadIdx.x & 31;
    const int wave = threadIdx.x >> 5;
    const int m  = lane & 15;    // M (A) / N (B,C) position within half-wave
    const int kk = lane >> 4;    // 0: K=0,1  1: K=2,3 within a K-step

    // Load B fragments for all 9 K-steps once (P is tiny & cached).
    v2f B[K_STEPS];
#pragma unroll
    for (int ks = 0; ks < K_STEPS; ++ks) {
        int a = ks * 4 + kk * 2;
        B[ks].x = Ppad[a * N_PAD + m];        // K = a,   N = m
        B[ks].y = Ppad[(a + 1) * N_PAD + m];  // K = a+1, N = m
    }

    const float* Fe = F + (size_t)e * (N_ALPHA * SOAP);
    float*       Oe = out + (size_t)e * (SOAP * N_J);

    // Each wave handles M-tiles [2*wave, 2*wave+1] -> 8 waves cover SOAP=256.
#pragma unroll
    for (int ti = 0; ti < 2; ++ti) {
        const int s0 = (wave * 2 + ti) * 16;
        v8f c = {};
#pragma unroll
        for (int ks = 0; ks < K_STEPS; ++ks) {
            const int a = ks * 4 + kk * 2;
            const float* p = Fe + a * SOAP + s0 + m;
            v2f av;
            av.x = __builtin_nontemporal_load(p);          // A[M=m, K=a]
            av.y = __builtin_nontemporal_load(p + SOAP);   // A[M=m, K=a+1]
            // D = A(16x4) * B(4x16) + C  -> v_wmma_f32_16x16x4_f32
            c = __builtin_amdgcn_wmma_f32_16x16x4_f32(
                    /*neg_a=*/false, av,
                    /*neg_b=*/false, B[ks],
                    /*c_mod=*/(short)0, c,
                    /*reuse_a=*/false, /*reuse_b=*/false);
        }
        // C layout: VGPR r, lanes 0-15 -> M=r, N=lane; lanes 16-31 -> M=r+8.
        if (m < N_J) {
#pragma unroll
            for (int r = 0; r < 8; ++r) {
                const int s = s0 + r + 8 * kk;
                __builtin_nontemporal_store(c[r], Oe + s * N_J + m);
            }
        }
    }
}

extern "C" void kernel_launch(void* const* d_in, const int* in_sizes, int n_in,
                              void* d_out, int out_size, void* d_ws, size_t ws_size,
                              hipStream_t stream) {
    const float* features = (const float*)d_in[0];
    const float* U        = (const float*)d_in[1];
    const int*   alpha1   = (const int*)d_in[2];
    const int*   alpha2   = (const int*)d_in[3];
    const int*   j1       = (const int*)d_in[4];
    const int*   j2       = (const int*)d_in[5];
    float*       out      = (float*)d_out;
    float*       Ppad     = (float*)d_ws;   // 36*16 floats

    const int env = in_sizes[0] / (N_ALPHA * SOAP);

    build_P_kernel<<<1, N_ALPHA * N_PAD, 0, stream>>>(U, alpha1, alpha2, j1, j2, Ppad);
    coupling_wmma_kernel<<<env, 256, 0, stream>>>(features, Ppad, out);
}